// Augmentor_70214125355069
// MI455X (gfx1250) — compile-verified
//
#include <hip/hip_runtime.h>
#include <stdint.h>

// out = x + err * z,  z = jax.random.normal(key(42), shape) reproduced exactly:
// Threefry-2x32-20 over counter pairs (p, p+n/2) with key (0,42), JAX uniform
// bit mapping to [nextafter(-1,0), 1), then sqrt(2)*erfinv (Giles polynomial,
// matching XLA's f32 ErfInv lowering).

typedef float v4f __attribute__((ext_vector_type(4)));

__device__ __forceinline__ uint32_t rotl32(uint32_t x, uint32_t r) {
    return (x << r) | (x >> (32u - r));
}

// Threefry-2x32, 20 rounds, key = (0, 42)  [jax.random.key(42) -> (hi,lo)=(0,42)]
__device__ __forceinline__ void threefry2x32_k42(uint32_t x0, uint32_t x1,
                                                 uint32_t& o0, uint32_t& o1) {
    const uint32_t k0 = 0u;
    const uint32_t k1 = 42u;
    const uint32_t k2 = k0 ^ k1 ^ 0x1BD11BDAu;

    x0 += k0; x1 += k1;

#define TF_ROUND(r) { x0 += x1; x1 = rotl32(x1, (r)); x1 ^= x0; }
    TF_ROUND(13) TF_ROUND(15) TF_ROUND(26) TF_ROUND(6)
    x0 += k1; x1 += k2 + 1u;
    TF_ROUND(17) TF_ROUND(29) TF_ROUND(16) TF_ROUND(24)
    x0 += k2; x1 += k0 + 2u;
    TF_ROUND(13) TF_ROUND(15) TF_ROUND(26) TF_ROUND(6)
    x0 += k0; x1 += k1 + 3u;
    TF_ROUND(17) TF_ROUND(29) TF_ROUND(16) TF_ROUND(24)
    x0 += k1; x1 += k2 + 4u;
    TF_ROUND(13) TF_ROUND(15) TF_ROUND(26) TF_ROUND(6)
    x0 += k2; x1 += k0 + 5u;
#undef TF_ROUND

    o0 = x0; o1 = x1;
}

// Giles single-precision erfinv (same coefficients as XLA ErfInv f32).
__device__ __forceinline__ float erfinv_f32(float x) {
    float w = -__logf((1.0f - x) * (1.0f + x));
    float p;
    if (w < 5.0f) {
        w -= 2.5f;
        p =              2.81022636e-08f;
        p = fmaf(p, w,   3.43273939e-07f);
        p = fmaf(p, w,  -3.5233877e-06f);
        p = fmaf(p, w,  -4.39150654e-06f);
        p = fmaf(p, w,   0.00021858087f);
        p = fmaf(p, w,  -0.00125372503f);
        p = fmaf(p, w,  -0.00417768164f);
        p = fmaf(p, w,   0.246640727f);
        p = fmaf(p, w,   1.50140941f);
    } else {
        w = __fsqrt_rn(w) - 3.0f;
        p =             -0.000200214257f;
        p = fmaf(p, w,   0.000100950558f);
        p = fmaf(p, w,   0.00134934322f);
        p = fmaf(p, w,  -0.00367342844f);
        p = fmaf(p, w,   0.00573950773f);
        p = fmaf(p, w,  -0.0076224613f);
        p = fmaf(p, w,   0.00943887047f);
        p = fmaf(p, w,   1.00167406f);
        p = fmaf(p, w,   2.83297682f);
    }
    return p * x;
}

// 32 random bits -> standard normal, exactly JAX's uniform->erfinv path.
__device__ __forceinline__ float bits_to_normal(uint32_t bits) {
    const float lo    = -0.99999994f;     // nextafterf(-1, 0)
    const float range = 1.0f - lo;        // RTNE constant-fold == JAX f32 (maxval-minval)
    uint32_t fb = (bits >> 9) | 0x3F800000u;
    float f = __uint_as_float(fb) - 1.0f; // uniform in [0, 1)
    float u = fmaxf(lo, fmaf(f, range, lo));
    return 1.41421356f * erfinv_f32(u);   // sqrt(2) * erfinv(u)
}

__global__ void __launch_bounds__(256)
augmentor_threefry_kernel(const float* __restrict__ x,
                          const float* __restrict__ e,
                          float* __restrict__ out,
                          unsigned int half) {
    unsigned int t  = blockIdx.x * blockDim.x + threadIdx.x;
    unsigned int p4 = t * 4u;

    if (p4 + 3u < half) {
        // Streaming b128 non-temporal loads: data is touched exactly once and
        // exceeds the 192 MB L2, so bypass temporal caching (TH=NT).
        v4f x0v = __builtin_nontemporal_load((const v4f*)(x + p4));
        v4f e0v = __builtin_nontemporal_load((const v4f*)(e + p4));
        v4f x1v = __builtin_nontemporal_load((const v4f*)(x + half + p4));
        v4f e1v = __builtin_nontemporal_load((const v4f*)(e + half + p4));

        v4f o0v, o1v;
#pragma unroll
        for (int i = 0; i < 4; ++i) {
            uint32_t r0, r1;
            // One 20-round Threefry yields TWO normals: elements p and p+half.
            threefry2x32_k42(p4 + (unsigned)i, half + p4 + (unsigned)i, r0, r1);
            o0v[i] = fmaf(e0v[i], bits_to_normal(r0), x0v[i]);
            o1v[i] = fmaf(e1v[i], bits_to_normal(r1), x1v[i]);
        }

        __builtin_nontemporal_store(o0v, (v4f*)(out + p4));
        __builtin_nontemporal_store(o1v, (v4f*)(out + half + p4));
    } else if (p4 < half) {
        // Scalar tail (not taken for the 2^26-element reference shape).
        for (unsigned int p = p4; p < half; ++p) {
            uint32_t r0, r1;
            threefry2x32_k42(p, half + p, r0, r1);
            out[p]        = fmaf(e[p],        bits_to_normal(r0), x[p]);
            out[half + p] = fmaf(e[half + p], bits_to_normal(r1), x[half + p]);
        }
    }
}

extern "C" void kernel_launch(void* const* d_in, const int* in_sizes, int n_in,
                              void* d_out, int out_size, void* d_ws, size_t ws_size,
                              hipStream_t stream) {
    (void)n_in; (void)out_size; (void)d_ws; (void)ws_size;

    const float* x = (const float*)d_in[0];      // xp_batch
    const float* e = (const float*)d_in[1];      // xp_err_batch
    float* out     = (float*)d_out;

    unsigned int n    = (unsigned int)in_sizes[0];  // 524288*128 = 2^26 (even)
    unsigned int half = n / 2u;                     // Threefry counter split point

    const unsigned int threads = 256;                        // 8 wave32 per block
    unsigned int pairs4 = (half + 3u) / 4u;                  // 4 pair-indices/thread
    unsigned int blocks = (pairs4 + threads - 1u) / threads; // 32768 for ref shape

    augmentor_threefry_kernel<<<blocks, threads, 0, stream>>>(x, e, out, half);
}